// UpMaxPooling_10033043604222
// MI455X (gfx1250) — compile-verified
//
#include <hip/hip_runtime.h>
#include <hip/hip_bf16.h>

// Max-unpooling scatter-add:
//   out = zeros(B*2H*2W*C); out[ind[i]] += x[i]  for all i
// Pure memory-bound scatter. No matrix math -> no WMMA applicable. CDNA5
// memory-path optimization: b128 non-temporal loads for the streamed
// input/index arrays, and native no-return global_atomic_add_f32 for the
// scatter (STOREcnt-tracked, no return latency).

typedef float     v4f __attribute__((ext_vector_type(4)));
typedef long long v2l __attribute__((ext_vector_type(2)));

// ---------------------------------------------------------------------------
// PROBE (never launched): single unsafeAtomicAdd, placed FIRST in the file so
// the disasm snippet shows exactly how the FP atomic lowers on gfx1250
// (expect: global_atomic_add_f32 no-return, not a cmpswap loop).
// ---------------------------------------------------------------------------
__global__ void atomic_probe(float* p, const float* v, const int* j) {
  unsafeAtomicAdd(p + j[threadIdx.x], v[threadIdx.x]);
}

// ---------------------------------------------------------------------------
// Scatter-add. 4 elements per thread:
//   1x global_load_b128 (x, NT) + 2x global_load_b128 (ind, NT)
//   4x no-return global_atomic_add_f32
// ---------------------------------------------------------------------------
__global__ __launch_bounds__(256) void scatter_add_kernel(
    const float* __restrict__ x,
    const long long* __restrict__ ind,
    float* __restrict__ out,
    int n) {
  const int base = (blockIdx.x * 256 + threadIdx.x) * 4;

  if (base + 3 < n) {
    v4f v  = __builtin_nontemporal_load((const v4f*)(x + base));
    v2l ja = __builtin_nontemporal_load((const v2l*)(ind + base));
    v2l jb = __builtin_nontemporal_load((const v2l*)(ind + base + 2));

    unsafeAtomicAdd(out + ja.x, v.x);
    unsafeAtomicAdd(out + ja.y, v.y);
    unsafeAtomicAdd(out + jb.x, v.z);
    unsafeAtomicAdd(out + jb.y, v.w);
  } else {
    // Tail (not hit for this problem size, but keep it correct).
    for (int i = base; i < n; ++i) {
      float v = x[i];
      long long j = ind[i];
      unsafeAtomicAdd(out + j, v);
    }
  }
}

// ---------------------------------------------------------------------------
// Zero-initialize the output (must happen every launch: the scatter
// accumulates, and the harness replays the graph without re-clearing d_out).
// Non-temporal b128 stores: pure streaming fill.
// ---------------------------------------------------------------------------
__global__ __launch_bounds__(256) void zero_out_kernel(float* __restrict__ out,
                                                       long long n) {
  const long long tid    = (long long)blockIdx.x * blockDim.x + threadIdx.x;
  const long long stride = (long long)gridDim.x * blockDim.x;

  v4f z = (v4f)0.0f;
  v4f* o4 = (v4f*)out;
  const long long n4 = n >> 2;

  for (long long k = tid; k < n4; k += stride) {
    __builtin_nontemporal_store(z, o4 + k);
  }
  // Scalar tail (n is a multiple of 4 for this problem, but be safe).
  for (long long k = (n4 << 2) + tid; k < n; k += stride) {
    __builtin_nontemporal_store(0.0f, out + k);
  }
}

extern "C" void kernel_launch(void* const* d_in, const int* in_sizes, int n_in,
                              void* d_out, int out_size, void* d_ws, size_t ws_size,
                              hipStream_t stream) {
  const float*     x   = (const float*)d_in[0];        // f32, B*H*W*C
  const long long* ind = (const long long*)d_in[1];    // int64 flat indices
  float*           out = (float*)d_out;                // f32, B*2H*2W*C

  const int       n     = in_sizes[0];                 // 33,554,432
  const long long total = (long long)out_size;         // 134,217,728

  // Phase 1: clear output. 512 MB of NT b128 stores; grid-stride with a
  // fixed large grid (8 wave32 waves per block).
  zero_out_kernel<<<4096, 256, 0, stream>>>(out, total);

  // Phase 2: scatter. One-shot grid, 4 elements per thread.
  const int threads_needed = (n + 3) / 4;
  const int blocks = (threads_needed + 255) / 256;
  scatter_add_kernel<<<blocks, 256, 0, stream>>>(x, ind, out, n);
}